// Embedding_65386582114481
// MI455X (gfx1250) — compile-verified
//
#include <hip/hip_runtime.h>

// Embedding gather: out[row, :] = weight[token_ids[row], :]
//   token_ids: int32[8192]  (4 x 2048 flattened)
//   weight:    f32[32000, 1024]
//   out:       f32[8192, 1024]
//
// Memory-bound: ~64 MB total traffic -> ~2.8 us at 23.3 TB/s HBM.
// One 256-thread block (8 wave32s) copies one 4 KB row: each lane moves
// exactly one float4 (global_load_b128 -> global_store_b128, NT hint on the
// store so the streamed output doesn't pollute the 192 MB L2 where gathered
// weight rows can be re-hit).

typedef float v4f __attribute__((ext_vector_type(4)));

#define EMB_DIM 1024
#define ROW_V4  (EMB_DIM / 4)   // 256 float4 per row == blockDim.x

__global__ __launch_bounds__(ROW_V4) void embedding_gather_kernel(
    const int* __restrict__ token_ids,
    const float* __restrict__ weight,
    float* __restrict__ out,
    int n_tokens)
{
    const int row = blockIdx.x;
    if (row >= n_tokens) return;

    // Uniform across the block -> compiler emits a scalar (SMEM) load,
    // keeping the vector-memory pipe dedicated to the 4 KB row payload.
    const int idx = token_ids[row];

    const v4f* __restrict__ src =
        (const v4f*)(weight + (size_t)idx * EMB_DIM);
    v4f* __restrict__ dst =
        (v4f*)(out + (size_t)row * EMB_DIM);

    const int t = threadIdx.x;           // 0..255, exactly one v4f each
    v4f v = src[t];                      // global_load_b128, regular temporal
                                         // (repeated token ids re-hit L2)
    __builtin_nontemporal_store(v, dst + t);  // global_store_b128 w/ NT hint
}

extern "C" void kernel_launch(void* const* d_in, const int* in_sizes, int n_in,
                              void* d_out, int out_size, void* d_ws, size_t ws_size,
                              hipStream_t stream)
{
    const int*   token_ids = (const int*)d_in[0];   // int32[B*S]
    const float* weight    = (const float*)d_in[1]; // f32[VOCAB, DIM]
    float*       out       = (float*)d_out;         // f32[B*S, DIM]

    const int n_tokens = in_sizes[0];               // 4 * 2048 = 8192

    dim3 grid(n_tokens);
    dim3 block(ROW_V4);                             // 256 threads = 8 wave32
    embedding_gather_kernel<<<grid, block, 0, stream>>>(token_ids, weight, out,
                                                        n_tokens);
}